// Distribution_Loss_43173011259409
// MI455X (gfx1250) — compile-verified
//
#include <hip/hip_runtime.h>

// ---------------------------------------------------------------------------
// Distribution NLL loss, elementwise over B rows.
//   out[i] = C + 0.5*log(ss) + log(y) + (log(y) - (mu1+mu2))^2 / (2*ss)
//   where ss = exp(2*ls1) + exp(2*ls2),  C = 0.5*log(2*pi)
// Memory-bound (AI ~0.6 FLOP/byte): optimize for pure streaming bandwidth.
// ---------------------------------------------------------------------------

typedef float v4f __attribute__((ext_vector_type(4)));

#define LOG_2PI_HALF 0.91893853320467274178f
#define LOG2E        1.44269504088896340736f   // 1/ln(2)
#define LN2          0.69314718055994530942f

// Single-instruction transcendentals: v_exp_f32 / v_log_f32 are base-2.
__device__ __forceinline__ float fast_exp(float x) {
    return __builtin_amdgcn_exp2f(x * LOG2E);
}
__device__ __forceinline__ float fast_log(float x) {
    return __builtin_amdgcn_logf(x) * LN2;
}

__device__ __forceinline__ float neg_log_f(float mu1, float ls1,
                                           float mu2, float ls2, float y) {
    // sigma1^2 + sigma2^2 = exp(2*ls1) + exp(2*ls2)
    float ss  = fast_exp(2.0f * ls1) + fast_exp(2.0f * ls2);
    float ly  = fast_log(y);
    float d   = ly - (mu1 + mu2);
    float inv = __builtin_amdgcn_rcpf(ss);            // v_rcp_f32
    return LOG_2PI_HALF + 0.5f * fast_log(ss) + ly + d * d * (0.5f * inv);
}

// Vector kernel: 4 rows per thread.
// Loads:  1 x b128 (truth), 4 x b128 (preds, 64 contiguous bytes/thread)
// Store:  1 x b128 (out)
// All traffic marked non-temporal: 234 MB single-pass stream > 192 MB L2.
__global__ __launch_bounds__(256) void dist_loss_vec4(
    const v4f* __restrict__ preds4,   // 4 floats per row -> one v4f per row
    const v4f* __restrict__ truth4,   // 4 rows of truth per v4f
    v4f* __restrict__ out4,           // 4 rows of output per v4f
    int n4)                           // number of 4-row groups
{
    int i = blockIdx.x * blockDim.x + threadIdx.x;
    if (i >= n4) return;

    v4f y  = __builtin_nontemporal_load(&truth4[i]);
    v4f p0 = __builtin_nontemporal_load(&preds4[4 * i + 0]);
    v4f p1 = __builtin_nontemporal_load(&preds4[4 * i + 1]);
    v4f p2 = __builtin_nontemporal_load(&preds4[4 * i + 2]);
    v4f p3 = __builtin_nontemporal_load(&preds4[4 * i + 3]);

    v4f r;
    r.x = neg_log_f(p0.x, p0.y, p0.z, p0.w, y.x);
    r.y = neg_log_f(p1.x, p1.y, p1.z, p1.w, y.y);
    r.z = neg_log_f(p2.x, p2.y, p2.z, p2.w, y.z);
    r.w = neg_log_f(p3.x, p3.y, p3.z, p3.w, y.w);

    __builtin_nontemporal_store(r, &out4[i]);
}

// Scalar tail for n % 4 != 0 (not hit for B = 8388608, kept for generality).
__global__ __launch_bounds__(64) void dist_loss_tail(
    const float* __restrict__ preds, const float* __restrict__ truth,
    float* __restrict__ out, int start, int n)
{
    int i = start + blockIdx.x * blockDim.x + threadIdx.x;
    if (i >= n) return;
    out[i] = neg_log_f(preds[4 * i + 0], preds[4 * i + 1],
                       preds[4 * i + 2], preds[4 * i + 3], truth[i]);
}

extern "C" void kernel_launch(void* const* d_in, const int* in_sizes, int n_in,
                              void* d_out, int out_size, void* d_ws, size_t ws_size,
                              hipStream_t stream) {
    const float* preds = (const float*)d_in[0];  // (B, 4) f32
    const float* truth = (const float*)d_in[1];  // (B, 1) f32
    float* out = (float*)d_out;                  // (B,)   f32

    const int n  = in_sizes[1];                  // B rows
    const int n4 = n >> 2;
    if (n4 > 0) {
        const int threads = 256;
        const int blocks  = (n4 + threads - 1) / threads;
        dist_loss_vec4<<<blocks, threads, 0, stream>>>(
            (const v4f*)preds, (const v4f*)truth, (v4f*)out, n4);
    }
    const int rem = n & 3;
    if (rem) {
        dist_loss_tail<<<1, 64, 0, stream>>>(preds, truth, out, n4 << 2, n);
    }
}